// MeshRefinementStage_84602265797176
// MI455X (gfx1250) — compile-verified
//
#include <hip/hip_runtime.h>
#include <math.h>

typedef __attribute__((ext_vector_type(2))) float v2f;
typedef __attribute__((ext_vector_type(8))) float v8f;
typedef __attribute__((ext_vector_type(4))) unsigned int u32x4;
typedef __attribute__((ext_vector_type(8))) int i32x8;
typedef __attribute__((ext_vector_type(4))) int i32x4;

#define HID 128
#define KP  132                          // K padded to multiple of 4 (131->132)
#define NPAIR (KP / 2)                   // 66 K-pairs
// Combined packed weights: row r = pair*2 + m  (m = matrix 0/1), 128 float2 per row
#define PWC_FLOATS (NPAIR * 2 * HID * 2) // 33792 floats
#define PWC_BYTES  (PWC_FLOATS * 4)      // 135168 bytes
#define LDS_BYTES  PWC_BYTES

__constant__ float c_limits[14] = {0.04f,0.06f,0.04f,0.04f,0.02f,0.02f,0.04f,
                                   0.04f,0.03f,0.03f,0.02f,0.02f,0.01f,0.01f};

// ---------------------------------------------------------------------------
// TDM: DMA a flat run of `bytes` (4B elements) from global into LDS.
// D# per CDNA5 ISA ch.8: group0 {count, lds_addr, global_addr, type=2},
// group1 {data_size=4B, tensor_dim0=tile_dim0=elems, dim1=1, stride=elems}.
// ---------------------------------------------------------------------------
__device__ __forceinline__ void tdm_load_to_lds(const void* gptr,
                                                unsigned lds_off,
                                                unsigned bytes) {
    unsigned long long ga = (unsigned long long)gptr;
    unsigned elems = bytes >> 2;                       // 4-byte elements
    u32x4 g0;
    g0[0] = 1u;                                        // count=1 (valid), user mode
    g0[1] = lds_off;                                   // lds_addr (bytes)
    g0[2] = (unsigned)(ga & 0xFFFFFFFFu);              // global_addr[31:0]
    g0[3] = (unsigned)((ga >> 32) & 0x01FFFFFFu)       // global_addr[56:32]
          | (2u << 30);                                // type=2 ("image")
    i32x8 g1;
    g1[0] = (int)(2u << 16);                           // data_size=2 -> 4B
    g1[1] = (int)((elems & 0xFFFFu) << 16);            // tensor_dim0[15:0]
    g1[2] = (int)(((elems >> 16) & 0xFFFFu)            // tensor_dim0[31:16]
          | (1u << 16));                               // tensor_dim1[15:0]=1
    g1[3] = (int)((elems & 0xFFFFu) << 16);            // tile_dim0 = elems
    g1[4] = 1;                                         // tile_dim1 = 1
    g1[5] = (int)elems;                                // tensor_dim0_stride lo
    g1[6] = 0;
    g1[7] = 0;
    i32x4 z4 = {0, 0, 0, 0};
    i32x8 z8 = {0, 0, 0, 0, 0, 0, 0, 0};
    __builtin_amdgcn_tensor_load_to_lds(g0, g1, z4, z4, z8, 0);
}

// feats = [normals | zeros] padded to KP columns
__global__ void k_init_feats(const float* __restrict__ normals,
                             float* __restrict__ feats, int N) {
    int t = blockIdx.x * blockDim.x + threadIdx.x;
    if (t >= N * KP) return;
    int n = t / KP, c = t - n * KP;
    feats[t] = (c < 3) ? normals[n * 3 + c] : 0.0f;
}

// Pack both layer matrices (d x 128, row major) into interleaved pair-major:
// pwc[((p*2 + m)*128 + n)*2 + {0,1}] = { Wm[2p][n], Wm[2p+1][n] }, zero pad.
__global__ void k_pack_w2(const float* __restrict__ w0,
                          const float* __restrict__ w1, int d,
                          float* __restrict__ pwc) {
    int t = blockIdx.x * blockDim.x + threadIdx.x;
    if (t >= NPAIR * 2 * HID) return;
    int p   = t / (2 * HID);
    int rem = t - p * 2 * HID;
    int m   = rem / HID;
    int n   = rem - m * HID;
    const float* w = m ? w1 : w0;
    float lo = (2 * p     < d) ? w[(2 * p)     * HID + n] : 0.0f;
    float hi = (2 * p + 1 < d) ? w[(2 * p + 1) * HID + n] : 0.0f;
    pwc[t * 2 + 0] = lo;
    pwc[t * 2 + 1] = hi;
}

// Dual GEMM: x0 = feats@W0 + b0 (into nbr buffer), x1 = feats@W1 + b1.
// Interleaved weights staged into LDS by one TDM op; all 8 B-fragments of a
// k-step sit in a 1.4KB window -> 4x ds_load_2addr_b64 off one address.
// Software-pipelined (double-buffered A + B frags), fully unrolled k loop.
// WMMA f32 16x16x4 lane layout: A: M=lane%16, K=2*(lane/16)+{0,1};
// B: N=lane%16, K=2*(lane/16)+{0,1}; C: M=r+8*(lane/16), N=lane%16.
__global__ __launch_bounds__(256)
void k_gemm_dual(const float* __restrict__ feats,
                 const float* __restrict__ pwc,
                 const float* __restrict__ b0, const float* __restrict__ b1,
                 float* __restrict__ x0_out, float* __restrict__ x1_out,
                 int N) {
    extern __shared__ float smem[];              // PWC_FLOATS

    if (threadIdx.x < 32) {                      // wave 0 drives the TDM
        tdm_load_to_lds(pwc, 0u, PWC_BYTES);
        __builtin_amdgcn_s_wait_tensorcnt(0);
    }
    __syncthreads();

    const int lane    = threadIdx.x & 31;
    const int wave    = threadIdx.x >> 5;
    const int rowTile = blockIdx.x * 4 + (wave >> 1);
    const int row0    = rowTile * 16;
    const int colBase = (wave & 1) * 64;
    const int mrow    = lane & 15;   // M index for A, N index for B/C
    const int kl      = lane >> 4;   // 0/1 -> K sub-offset 0/2

    const float* arow  = feats + (size_t)(row0 + mrow) * KP + 2 * kl;
    // combined row for (k0, m): r = k0 + 2*kl + m ; float2 [r*HID + col]
    const float* bbase = smem + ((size_t)(2 * kl) * HID + colBase + mrow) * 2;

    v8f acc[2][4];
#pragma unroll
    for (int m = 0; m < 2; ++m)
#pragma unroll
        for (int ct = 0; ct < 4; ++ct)
            acc[m][ct] = (v8f){0.f,0.f,0.f,0.f,0.f,0.f,0.f,0.f};

    v2f a_cur, a_nxt;
    v2f bf_cur[2][4], bf_nxt[2][4];

    a_cur = *(const v2f*)(arow);
#pragma unroll
    for (int m = 0; m < 2; ++m)
#pragma unroll
        for (int ct = 0; ct < 4; ++ct)
            bf_cur[m][ct] = *(const v2f*)(bbase + (m * HID + ct * 16) * 2);

#pragma unroll
    for (int k0 = 0; k0 < KP; k0 += 4) {
        if (k0 + 4 < KP) {
            a_nxt = *(const v2f*)(arow + k0 + 4);
            const float* bp = bbase + (size_t)(k0 + 4) * HID * 2;
#pragma unroll
            for (int m = 0; m < 2; ++m)
#pragma unroll
                for (int ct = 0; ct < 4; ++ct)
                    bf_nxt[m][ct] = *(const v2f*)(bp + (m * HID + ct * 16) * 2);
        }
#pragma unroll
        for (int ct = 0; ct < 4; ++ct) {
            acc[0][ct] = __builtin_amdgcn_wmma_f32_16x16x4_f32(
                false, a_cur, false, bf_cur[0][ct], (short)0, acc[0][ct],
                false, false);
            acc[1][ct] = __builtin_amdgcn_wmma_f32_16x16x4_f32(
                false, a_cur, false, bf_cur[1][ct], (short)0, acc[1][ct],
                false, false);
        }
        a_cur = a_nxt;
#pragma unroll
        for (int m = 0; m < 2; ++m)
#pragma unroll
            for (int ct = 0; ct < 4; ++ct)
                bf_cur[m][ct] = bf_nxt[m][ct];
    }

#pragma unroll
    for (int ct = 0; ct < 4; ++ct) {
        int col = colBase + ct * 16 + mrow;
        float bias0 = b0[col];
        float bias1 = b1[col];
#pragma unroll
        for (int r = 0; r < 8; ++r) {
            int row = row0 + r + 8 * kl;
            x0_out[(size_t)row * HID + col] = acc[0][ct][r] + bias0;
            x1_out[(size_t)row * HID + col] = acc[1][ct][r] + bias1;
        }
    }
}

// nbr[e0] += x1[e1]; nbr[e1] += x1[e0]  (float atomics, 4 cols per thread)
__global__ void k_scatter(const int* __restrict__ edges,
                          const float* __restrict__ x1,
                          float* __restrict__ nbr, int E) {
    int t = blockIdx.x * blockDim.x + threadIdx.x;
    int e = t >> 5;
    if (e >= E) return;
    int cg = (t & 31) * 4;
    int e0 = edges[2 * e + 0];
    int e1 = edges[2 * e + 1];
    float4 va = *(const float4*)(x1 + (size_t)e1 * HID + cg);
    float4 vb = *(const float4*)(x1 + (size_t)e0 * HID + cg);
    float* p0 = nbr + (size_t)e0 * HID + cg;
    float* p1 = nbr + (size_t)e1 * HID + cg;
    atomicAdd(p0 + 0, va.x); atomicAdd(p0 + 1, va.y);
    atomicAdd(p0 + 2, va.z); atomicAdd(p0 + 3, va.w);
    atomicAdd(p1 + 0, vb.x); atomicAdd(p1 + 1, vb.y);
    atomicAdd(p1 + 2, vb.z); atomicAdd(p1 + 3, vb.w);
}

// feats = [relu(nbr) | normals | 0]; optionally emit h = relu(nbr) to output.
__global__ void k_fuse(const float* __restrict__ nbr,
                       const float* __restrict__ normals,
                       float* __restrict__ feats,
                       float* __restrict__ h_out, int N) {
    int t = blockIdx.x * blockDim.x + threadIdx.x;
    if (t >= N * KP) return;
    int n = t / KP, c = t - n * KP;
    float v;
    if (c < HID) {
        v = nbr[(size_t)n * HID + c];
        v = v > 0.0f ? v : 0.0f;
        if (h_out) h_out[(size_t)n * HID + c] = v;
    } else if (c < HID + 3) {
        v = normals[n * 3 + (c - HID)];
    } else {
        v = 0.0f;
    }
    feats[t] = v;
}

// One wave per vertex: deform = tanh(feats @ w_off + b_off), clamp by part
// limits, new_verts = verts + clamp * anchor.
__global__ __launch_bounds__(256)
void k_deform(const float* __restrict__ feats,
              const float* __restrict__ w_off, const float* __restrict__ b_off,
              const float* __restrict__ verts,
              const float* __restrict__ anchor_v,
              const int* __restrict__ part_id,
              float* __restrict__ out, int N, int V) {
    int wv   = (blockIdx.x * blockDim.x + threadIdx.x) >> 5;
    int lane = threadIdx.x & 31;
    if (wv >= N) return;
    const float* frow = feats + (size_t)wv * KP;
    float a0 = 0.f, a1 = 0.f, a2 = 0.f;
    for (int k = lane; k < HID + 3; k += 32) {
        float f = frow[k];
        a0 += f * w_off[k * 3 + 0];
        a1 += f * w_off[k * 3 + 1];
        a2 += f * w_off[k * 3 + 2];
    }
    for (int off = 16; off > 0; off >>= 1) {
        a0 += __shfl_down(a0, off, 32);
        a1 += __shfl_down(a1, off, 32);
        a2 += __shfl_down(a2, off, 32);
    }
    if (lane == 0) {
        int vv = wv % V;
        float lim = c_limits[part_id[vv]];
        float d[3] = {a0 + b_off[0], a1 + b_off[1], a2 + b_off[2]};
#pragma unroll
        for (int j = 0; j < 3; ++j) {
            float c = tanhf(d[j]);
            c = fminf(fmaxf(c, -lim), lim);
            out[(size_t)wv * 3 + j] =
                verts[(size_t)wv * 3 + j] + c * anchor_v[vv * 3 + j];
        }
    }
}

extern "C" void kernel_launch(void* const* d_in, const int* in_sizes, int n_in,
                              void* d_out, int out_size, void* d_ws,
                              size_t ws_size, hipStream_t stream) {
    (void)n_in; (void)out_size; (void)ws_size;

    const float* verts    = (const float*)d_in[0];
    const float* normals  = (const float*)d_in[1];
    const float* anchor_v = (const float*)d_in[2];
    const int*   edges    = (const int*)d_in[3];
    const int*   part_id  = (const int*)d_in[4];
    const float* w[3][2]  = {{(const float*)d_in[5],  (const float*)d_in[7]},
                             {(const float*)d_in[9],  (const float*)d_in[11]},
                             {(const float*)d_in[13], (const float*)d_in[15]}};
    const float* bias[3][2] = {{(const float*)d_in[6],  (const float*)d_in[8]},
                               {(const float*)d_in[10], (const float*)d_in[12]},
                               {(const float*)d_in[14], (const float*)d_in[16]}};
    const int dlayer[3] = {3, HID + 3, HID + 3};
    const float* w_off = (const float*)d_in[17];
    const float* b_off = (const float*)d_in[18];

    const int N = in_sizes[0] / 3;   // 211072
    const int V = in_sizes[2] / 3;   // 6596
    const int E = in_sizes[3] / 2;   // 633216

    // Workspace carve-out (256B aligned)
    char*  ws  = (char*)d_ws;
    size_t off = 0;
    auto carve = [&](size_t bytes) -> void* {
        void* p = ws + off;
        off = (off + bytes + 255) & ~(size_t)255;
        return p;
    };
    float* feats = (float*)carve((size_t)N * KP * sizeof(float));
    float* x1    = (float*)carve((size_t)N * HID * sizeof(float));
    float* nbr   = (float*)carve((size_t)N * HID * sizeof(float));
    float* pwc[3];
    for (int l = 0; l < 3; ++l)
        pwc[l] = (float*)carve((size_t)PWC_BYTES);

    // Allow 135KB dynamic LDS on the GEMM kernel (non-stream call: capture-safe)
    (void)hipFuncSetAttribute((const void*)k_gemm_dual,
                              hipFuncAttributeMaxDynamicSharedMemorySize,
                              LDS_BYTES);

    const int T = 256;

    k_init_feats<<<((size_t)N * KP + T - 1) / T, T, 0, stream>>>(normals, feats, N);

    for (int l = 0; l < 3; ++l)
        k_pack_w2<<<(NPAIR * 2 * HID + T - 1) / T, T, 0, stream>>>(
            w[l][0], w[l][1], dlayer[l], pwc[l]);

    float* h_out_final = (float*)d_out + (size_t)N * 3;

    for (int l = 0; l < 3; ++l) {
        // x0 (+bias) lands directly in nbr; scatter accumulates on top.
        k_gemm_dual<<<N / 64, T, LDS_BYTES, stream>>>(
            feats, pwc[l], bias[l][0], bias[l][1], nbr, x1, N);
        k_scatter<<<((size_t)E * 32 + T - 1) / T, T, 0, stream>>>(edges, x1,
                                                                  nbr, E);
        k_fuse<<<((size_t)N * KP + T - 1) / T, T, 0, stream>>>(
            nbr, normals, feats, (l == 2) ? h_out_final : nullptr, N);
    }

    k_deform<<<((size_t)N * 32 + T - 1) / T, T, 0, stream>>>(
        feats, w_off, b_off, verts, anchor_v, part_id, (float*)d_out, N, V);
}